// VendiModule_74036646249283
// MI455X (gfx1250) — compile-verified
//
#include <hip/hip_runtime.h>
#include <hip/hip_bf16.h>

// Problem constants (match reference)
#define R_DIM 2048            // rows / kernel-matrix dim
#define P_DIM 4096            // points per row
#define D_DIM (P_DIM * 3)     // 12288 features
#define NTERMS 32             // entropy series terms: -x ln x = sum_m x(1-x)^m / m

// GEMM tiling
#define KC 32                 // k-chunk staged in LDS
#define LDS_STRIDE 36         // 32 + 4 pad (16B aligned, kills bank conflicts)

typedef __attribute__((ext_vector_type(2))) float v2f;  // A/B frag of wmma f32 16x16x4
typedef __attribute__((ext_vector_type(8))) float v8f;  // C/D frag

// GCC-style vector to match the async builtin's parameter type exactly
typedef int b128_t __attribute__((vector_size(16)));
#define AS_GLOBAL __attribute__((address_space(1)))
#define AS_SHARED __attribute__((address_space(3)))

// CDNA5 async global->LDS path (ASYNCcnt), guarded so we always compile.
#if __has_builtin(__builtin_amdgcn_global_load_async_to_lds_b128) && \
    __has_builtin(__builtin_amdgcn_s_wait_asynccnt)
#define USE_ASYNC_LDS 1
#else
#define USE_ASYNC_LDS 0
#endif

// ---------------------------------------------------------------------------
// Stage 1: per-row invariant transform. One block per row r.
// ---------------------------------------------------------------------------
__global__ __launch_bounds__(256) void invariant_kernel(
    const float* __restrict__ pos, float* __restrict__ X, float* __restrict__ norms)
{
    const int r   = blockIdx.x;
    const int tid = threadIdx.x;
    const float* prow = pos + (size_t)r * P_DIM * 3;

    __shared__ float red[3 * 256];
    __shared__ float Ush[9];
    __shared__ float comsh[3];

    float sx = 0.f, sy = 0.f, sz = 0.f;
    for (int p = tid; p < P_DIM; p += 256) {
        sx += prow[p * 3 + 0];
        sy += prow[p * 3 + 1];
        sz += prow[p * 3 + 2];
    }
    red[tid] = sx; red[256 + tid] = sy; red[512 + tid] = sz;
    __syncthreads();
    for (int s = 128; s > 0; s >>= 1) {
        if (tid < s) {
            red[tid]       += red[tid + s];
            red[256 + tid] += red[256 + tid + s];
            red[512 + tid] += red[512 + tid + s];
        }
        __syncthreads();
    }

    if (tid == 0) {
        const float invP = 1.0f / (float)P_DIM;
        const float cx = red[0] * invP, cy = red[256] * invP, cz = red[512] * invP;
        comsh[0] = cx; comsh[1] = cy; comsh[2] = cz;

        float tx = prow[(P_DIM - 1) * 3 + 0] - cx;
        float ty = prow[(P_DIM - 1) * 3 + 1] - cy;
        float tz = prow[(P_DIM - 1) * 3 + 2] - cz;
        float na = sqrtf(tx * tx + ty * ty + tz * tz);
        float ax = tx / na, ay = ty / na, az = tz / na;

        float dot = ax;                       // A . (1,0,0)
        float cn  = sqrtf(ay * ay + az * az); // |A x (1,0,0)|

        float Gm[3][3] = { { dot,  cn, 0.f },
                           { -cn, dot, 0.f },
                           { 0.f, 0.f, 1.f } };

        float vx = 1.0f - dot * ax, vy = -dot * ay, vz = -dot * az;
        float nv = sqrtf(vx * vx + vy * vy + vz * vz);
        vx /= nv; vy /= nv; vz /= nv;
        float wx = 0.f, wy = -az, wz = ay;    // w = B x A

        float Fi[3][3] = { { ax, vx, wx },
                           { ay, vy, wy },
                           { az, vz, wz } };
        float det = Fi[0][0] * (Fi[1][1] * Fi[2][2] - Fi[1][2] * Fi[2][1])
                  - Fi[0][1] * (Fi[1][0] * Fi[2][2] - Fi[1][2] * Fi[2][0])
                  + Fi[0][2] * (Fi[1][0] * Fi[2][1] - Fi[1][1] * Fi[2][0]);
        float id = 1.0f / det;
        float F[3][3];
        F[0][0] =  (Fi[1][1] * Fi[2][2] - Fi[1][2] * Fi[2][1]) * id;
        F[0][1] = -(Fi[0][1] * Fi[2][2] - Fi[0][2] * Fi[2][1]) * id;
        F[0][2] =  (Fi[0][1] * Fi[1][2] - Fi[0][2] * Fi[1][1]) * id;
        F[1][0] = -(Fi[1][0] * Fi[2][2] - Fi[1][2] * Fi[2][0]) * id;
        F[1][1] =  (Fi[0][0] * Fi[2][2] - Fi[0][2] * Fi[2][0]) * id;
        F[1][2] = -(Fi[0][0] * Fi[1][2] - Fi[0][2] * Fi[1][0]) * id;
        F[2][0] =  (Fi[1][0] * Fi[2][1] - Fi[1][1] * Fi[2][0]) * id;
        F[2][1] = -(Fi[0][0] * Fi[2][1] - Fi[0][1] * Fi[2][0]) * id;
        F[2][2] =  (Fi[0][0] * Fi[1][1] - Fi[0][1] * Fi[1][0]) * id;

        float T[3][3];
        for (int i = 0; i < 3; ++i)
            for (int j = 0; j < 3; ++j)
                T[i][j] = Fi[i][0] * Gm[0][j] + Fi[i][1] * Gm[1][j] + Fi[i][2] * Gm[2][j];
        for (int i = 0; i < 3; ++i)
            for (int j = 0; j < 3; ++j)
                Ush[i * 3 + j] = T[i][0] * F[0][j] + T[i][1] * F[1][j] + T[i][2] * F[2][j];
    }
    __syncthreads();

    const float cx = comsh[0], cy = comsh[1], cz = comsh[2];
    const float U00 = Ush[0], U01 = Ush[1], U02 = Ush[2];
    const float U10 = Ush[3], U11 = Ush[4], U12 = Ush[5];
    const float U20 = Ush[6], U21 = Ush[7], U22 = Ush[8];

    float* xrow = X + (size_t)r * D_DIM;
    float acc = 0.f;
    for (int p = tid; p < P_DIM; p += 256) {
        float x = prow[p * 3 + 0] - cx;
        float y = prow[p * 3 + 1] - cy;
        float z = prow[p * 3 + 2] - cz;
        float o0 = x * U00 + y * U10 + z * U20;
        float o1 = x * U01 + y * U11 + z * U21;
        float o2 = x * U02 + y * U12 + z * U22;
        xrow[p * 3 + 0] = o0;
        xrow[p * 3 + 1] = o1;
        xrow[p * 3 + 2] = o2;
        acc += o0 * o0 + o1 * o1 + o2 * o2;
    }
    __syncthreads();
    red[tid] = acc;
    __syncthreads();
    for (int s = 128; s > 0; s >>= 1) {
        if (tid < s) red[tid] += red[tid + s];
        __syncthreads();
    }
    if (tid == 0) norms[r] = red[0];
}

// ---------------------------------------------------------------------------
// Unified WMMA fp32 GEMM.
//   TRANSB=true : C = A * B^T  (gram; B rows are feature rows, transposed read)
//   TRANSB=false: C = A * B    (series matmul)
// Block = 256 threads = 8 waves (4x2 tiles of 16x16) -> 64x32 output region.
// Double-buffered LDS; async global->LDS copies one chunk ahead when available.
// ---------------------------------------------------------------------------
template <int KDIM, bool TRANSB>
__global__ __launch_bounds__(256) void wmma_gemm_kernel(
    const float* __restrict__ A, const float* __restrict__ B, float* __restrict__ C)
{
    __shared__ float lds[2 * 96 * LDS_STRIDE];
    const int tid   = threadIdx.x;
    const int wave  = tid >> 5;
    const int lane  = tid & 31;
    const int ibase = blockIdx.x * 64;
    const int jbase = blockIdx.y * 32;
    const int ti    = (wave >> 1) * 16;
    const int tj    = (wave & 1) * 16;
    const int lrow  = lane & 15;
    const int khalf = (lane >> 4) * 2;   // lanes 0-15 -> K{0,1}; 16-31 -> K{2,3}

    // This thread's 3 staging slots: idx = tid + it*256 -> (row, col4)
    auto src_ptr = [&](int it, int k0) -> const float* {
        int idx = tid + it * 256;
        int row = idx >> 3;
        int c4  = (idx & 7) * 4;
        if (row < 64) return A + (size_t)(ibase + row) * KDIM + k0 + c4;
        if constexpr (TRANSB)
            return B + (size_t)(jbase + row - 64) * KDIM + k0 + c4;      // B rows (n) x k
        else
            return B + (size_t)(k0 + row - 64) * KDIM + jbase + c4;     // B rows (k) x n
    };
    auto dst_ptr = [&](int it, int buf) -> float* {
        int idx = tid + it * 256;
        int row = idx >> 3;
        int c4  = (idx & 7) * 4;
        return lds + buf * (96 * LDS_STRIDE) + row * LDS_STRIDE + c4;
    };

    auto stage = [&](int k0, int buf) {
#if USE_ASYNC_LDS
        #pragma unroll
        for (int it = 0; it < 3; ++it) {
            // param0: global src (device AS, int4*), param1: LDS dst (AS3, int4*)
            AS_GLOBAL b128_t* gsrc =
                (AS_GLOBAL b128_t*)(uintptr_t)src_ptr(it, k0);
            AS_SHARED b128_t* ldst =
                (AS_SHARED b128_t*)(uint32_t)(uintptr_t)dst_ptr(it, buf);
            __builtin_amdgcn_global_load_async_to_lds_b128(gsrc, ldst, 0, 0);
        }
#else
        float4 v0 = *(const float4*)src_ptr(0, k0);
        float4 v1 = *(const float4*)src_ptr(1, k0);
        float4 v2 = *(const float4*)src_ptr(2, k0);
        if (k0 + KC < KDIM) {
            __builtin_prefetch((const void*)src_ptr(0, k0 + KC), 0, 0);
            __builtin_prefetch((const void*)src_ptr(1, k0 + KC), 0, 0);
            __builtin_prefetch((const void*)src_ptr(2, k0 + KC), 0, 0);
        }
        float* d0 = dst_ptr(0, buf); d0[0] = v0.x; d0[1] = v0.y; d0[2] = v0.z; d0[3] = v0.w;
        float* d1 = dst_ptr(1, buf); d1[0] = v1.x; d1[1] = v1.y; d1[2] = v1.z; d1[3] = v1.w;
        float* d2 = dst_ptr(2, buf); d2[0] = v2.x; d2[1] = v2.y; d2[2] = v2.z; d2[3] = v2.w;
#endif
    };

    v8f c = {};
    constexpr int NCHUNK = KDIM / KC;

    stage(0, 0);
    for (int ch = 0; ch < NCHUNK; ++ch) {
        const int buf = ch & 1;
        const bool has_next = (ch + 1) < NCHUNK;
        if (has_next) stage((ch + 1) * KC, buf ^ 1);
#if USE_ASYNC_LDS
        if (has_next) __builtin_amdgcn_s_wait_asynccnt(3);  // current chunk landed
        else          __builtin_amdgcn_s_wait_asynccnt(0);
#endif
        __syncthreads();

        const float* base = lds + buf * (96 * LDS_STRIDE);
        const float* arow = base + (ti + lrow) * LDS_STRIDE;
        #pragma unroll
        for (int kk = 0; kk < KC; kk += 4) {
            v2f a, b;
            a.x = arow[kk + khalf];
            a.y = arow[kk + khalf + 1];
            if constexpr (TRANSB) {
                const float* brow = base + (64 + tj + lrow) * LDS_STRIDE;  // B[k][n]=Xj[n][k]
                b.x = brow[kk + khalf];
                b.y = brow[kk + khalf + 1];
            } else {
                const float* bb = base + 64 * LDS_STRIDE;
                const int n = tj + lrow;
                b.x = bb[(kk + khalf)     * LDS_STRIDE + n];
                b.y = bb[(kk + khalf + 1) * LDS_STRIDE + n];
            }
            c = __builtin_amdgcn_wmma_f32_16x16x4_f32(
                    false, a, false, b, (short)0, c, false, false);
        }
        __syncthreads();
    }

    // C/D layout: lane<16 -> M=r, N=lane ; lane>=16 -> M=8+r, N=lane-16
    const int row0 = ibase + ti + (lane >> 4) * 8;
    const int col  = jbase + tj + (lane & 15);
    #pragma unroll
    for (int rr = 0; rr < 8; ++rr)
        C[(size_t)(row0 + rr) * R_DIM + col] = c[rr];
}

// ---------------------------------------------------------------------------
// K_ = exp(-max(n_i + n_j - 2 g, 0)) / R (in place over G); Bm = I - K_
// ---------------------------------------------------------------------------
__global__ __launch_bounds__(256) void kbuild_kernel(
    float* __restrict__ G, const float* __restrict__ nrm, float* __restrict__ Bm)
{
    size_t idx = (size_t)blockIdx.x * 256 + threadIdx.x;
    int i = (int)(idx / R_DIM);
    int j = (int)(idx % R_DIM);
    float g  = G[idx];
    float sq = fmaxf(nrm[i] + nrm[j] - 2.0f * g, 0.0f);
    float k_ = expf(-sq) * (1.0f / (float)R_DIM);   // GAMMA = 1
    G[idx]  = k_;
    Bm[idx] = ((i == j) ? 1.0f : 0.0f) - k_;
}

// ---------------------------------------------------------------------------
__global__ __launch_bounds__(256) void trace_kernel(
    const float* __restrict__ M, float* __restrict__ out)
{
    __shared__ float red[256];
    int tid = threadIdx.x;
    float s = 0.f;
    for (int i = tid; i < R_DIM; i += 256) s += M[(size_t)i * R_DIM + i];
    red[tid] = s;
    __syncthreads();
    for (int st = 128; st > 0; st >>= 1) {
        if (tid < st) red[tid] += red[tid + st];
        __syncthreads();
    }
    if (tid == 0) *out = red[0];
}

// entropy = sum_m tr(K_ (I-K_)^m)/m ;  output = -NU * entropy
__global__ void final_kernel(const float* __restrict__ T, float* __restrict__ out)
{
    if (threadIdx.x == 0 && blockIdx.x == 0) {
        float e = 0.f;
        for (int m = 1; m <= NTERMS; ++m) e += T[m] / (float)m;
        out[0] = -e;
    }
}

// ---------------------------------------------------------------------------
extern "C" void kernel_launch(void* const* d_in, const int* in_sizes, int n_in,
                              void* d_out, int out_size, void* d_ws, size_t ws_size,
                              hipStream_t stream)
{
    (void)in_sizes; (void)n_in; (void)out_size; (void)ws_size;
    const float* pos = (const float*)d_in[0];

    float* ws  = (float*)d_ws;
    float* X   = ws;                                   // 2048*12288
    float* G   = X  + (size_t)R_DIM * D_DIM;           // 2048^2 (K_ / M ping)
    float* Bm  = G  + (size_t)R_DIM * R_DIM;           // 2048^2 (I - K_)
    float* Mp  = Bm + (size_t)R_DIM * R_DIM;           // 2048^2 (M pong)
    float* nrm = Mp + (size_t)R_DIM * R_DIM;           // 2048
    float* Tb  = nrm + R_DIM;                          // NTERMS+1

    // Stage 1: invariant features + row norms
    invariant_kernel<<<R_DIM, 256, 0, stream>>>(pos, X, nrm);

    // Stage 2: Gram = X X^T (WMMA fp32, async-LDS double buffered)
    wmma_gemm_kernel<D_DIM, true>
        <<<dim3(R_DIM / 64, R_DIM / 32), 256, 0, stream>>>(X, X, G);

    // Stage 3: K_ (in G) and B = I - K_
    kbuild_kernel<<<(int)(((size_t)R_DIM * R_DIM) / 256), 256, 0, stream>>>(G, nrm, Bm);

    // Stage 4: entropy series. M_m = M_{m-1} * B, T_m = tr(M_m), M_0 = K_.
    float* cur = G;
    float* nxt = Mp;
    for (int m = 1; m <= NTERMS; ++m) {
        wmma_gemm_kernel<R_DIM, false>
            <<<dim3(R_DIM / 64, R_DIM / 32), 256, 0, stream>>>(cur, Bm, nxt);
        trace_kernel<<<1, 256, 0, stream>>>(nxt, Tb + m);
        float* t = cur; cur = nxt; nxt = t;
    }

    final_kernel<<<1, 1, 0, stream>>>(Tb, (float*)d_out);
}